// Reverb_27659589386409
// MI455X (gfx1250) — compile-verified
//
#include <hip/hip_runtime.h>

typedef __attribute__((ext_vector_type(2))) float v2f;
typedef __attribute__((ext_vector_type(8))) float v8f;

#define SDIM 256
#define LTOT (SDIM * SDIM)
#define NC 64
#define NCO 64
#define TILE_L 32
#define H2_STRIDE 40   // 2*40 mod 64 = 16 -> half-wave bank windows disjoint
#define CV_STRIDE 72   // 2*72 mod 64 = 16 -> half-wave bank windows disjoint

__global__ __launch_bounds__(256) void reverb_fused(
    const float* __restrict__ x,
    const float* __restrict__ weight,
    const float* __restrict__ conv,
    float* __restrict__ out)
{
    __shared__ float convs[NC * CV_STRIDE];   // padded 64 x 64 (A = conv^T)
    __shared__ float h2s[NC * H2_STRIDE];     // padded 64 x 32 (B tiles)

    const int t    = threadIdx.x;
    const int lane = t & 31;
    const int wv   = t >> 5;

    const int l0   = blockIdx.x * TILE_L;     // 32 positions, all in one image row
    const int r    = l0 >> 8;
    const int col0 = l0 & (SDIM - 1);

    // ---- stage conv (C x CO, row-major c*64+o) into padded LDS ----
#pragma unroll
    for (int i = 0; i < 16; ++i) {
        int idx = t + i * 256;
        convs[(idx >> 6) * CV_STRIDE + (idx & 63)] = conv[idx];
    }

    // ---- phase 1: h2[c][l] = x[c][l] * sum of 9 shifted weights (edge-masked) ----
    {
        const int col = col0 + lane;
        const int l   = l0 + lane;
        const bool up = (r > 0);
        const bool dn = (r < SDIM - 1);
        const bool lf = (col > 0);
        const bool rt = (col < SDIM - 1);
#pragma unroll
        for (int it = 0; it < 8; ++it) {
            const int c = wv * 8 + it;                       // wave owns 8 channels
            const float xv = x[(size_t)c * LTOT + l];
            const float* wb = weight + (size_t)c * 9 * LTOT + l;
            float wsum = 0.f;
            if (up) {                                        // di = -1  (tap i = 2)
                if (lf) wsum += wb[(size_t)8 * LTOT - SDIM - 1];
                        wsum += wb[(size_t)7 * LTOT - SDIM];
                if (rt) wsum += wb[(size_t)6 * LTOT - SDIM + 1];
            }
            {                                                // di = 0   (tap i = 1)
                if (lf) wsum += wb[(size_t)5 * LTOT - 1];
                        wsum += wb[(size_t)4 * LTOT];
                if (rt) wsum += wb[(size_t)3 * LTOT + 1];
            }
            if (dn) {                                        // di = +1  (tap i = 0)
                if (lf) wsum += wb[(size_t)2 * LTOT + SDIM - 1];
                        wsum += wb[(size_t)1 * LTOT + SDIM];
                if (rt) wsum += wb[SDIM + 1];
            }
            h2s[c * H2_STRIDE + lane] = xv * wsum;
        }
    }

    __syncthreads();

    // ---- phase 2: per-wave 16(CO) x 16(L) tile, K=64 in 16 WMMA f32 16x16x4 steps ----
    const int m_tile = wv & 3;       // CO chunk
    const int n_tile = wv >> 2;      // L chunk within block tile
    const int l15 = lane & 15;
    const int hi  = lane >> 4;       // half-wave: K pair base 0 or 2
    const int mrow = m_tile * 16 + l15;

    // A = conv^T : A[o, c] = conv[c*64 + o]; 16x4 f32 frag = 2 VGPRs
    v2f afrag[16];
#pragma unroll
    for (int kc = 0; kc < 16; ++kc) {
        const int c0 = kc * 4 + 2 * hi;
        afrag[kc].x = convs[c0 * CV_STRIDE + mrow];          // K = c0
        afrag[kc].y = convs[(c0 + 1) * CV_STRIDE + mrow];    // K = c0+1
    }

    v8f acc = {};
#pragma unroll
    for (int kc = 0; kc < 16; ++kc) {
        const int c0 = kc * 4 + 2 * hi;
        v2f b;
        b.x = h2s[c0 * H2_STRIDE + n_tile * 16 + l15];       // B[K=c0,   N]
        b.y = h2s[(c0 + 1) * H2_STRIDE + n_tile * 16 + l15]; // B[K=c0+1, N]
        acc = __builtin_amdgcn_wmma_f32_16x16x4_f32(
            false, afrag[kc], false, b, (short)0, acc, false, false);
    }

    // ---- store D: VGPR v <-> M = v + 8*hi ; lane%16 <-> N ----
    const int lout = l0 + n_tile * 16 + l15;
#pragma unroll
    for (int v = 0; v < 8; ++v) {
        const int o = m_tile * 16 + v + 8 * hi;
        out[(size_t)o * LTOT + lout] = acc[v];
    }
}

extern "C" void kernel_launch(void* const* d_in, const int* in_sizes, int n_in,
                              void* d_out, int out_size, void* d_ws, size_t ws_size,
                              hipStream_t stream) {
    const float* x      = (const float*)d_in[0];   // (1, 64, 256, 256)
    const float* weight = (const float*)d_in[1];   // (1, 576, 65536)
    const float* conv   = (const float*)d_in[2];   // (64, 64)
    float* out          = (float*)d_out;           // (1, 64, 256, 256)

    reverb_fused<<<dim3(LTOT / TILE_L), dim3(256), 0, stream>>>(x, weight, conv, out);
}